// YOLO3Loss_81698867905060
// MI455X (gfx1250) — compile-verified
//
#include <hip/hip_runtime.h>
#include <hip/hip_bf16.h>
#include <math.h>

#define NB 64
#define NT 50
#define TOTAL 10647
#define NUM_CLASSES 80
#define PRED_C 85
#define IMG_SIZE 416.0f
#define IGNORE_THR 0.5f
#define EPSV 1e-7f

typedef __attribute__((ext_vector_type(2))) float v2f;
typedef __attribute__((ext_vector_type(8))) float v8f;

__device__ __constant__ float c_fm[3]  = {13.f, 26.f, 52.f};
__device__ __constant__ int   c_off[3] = {0, 507, 2535};
__device__ __constant__ float c_anc[3][3][2] = {
    {{3.625f, 2.8125f}, {4.875f, 6.1875f}, {11.65625f, 10.1875f}},
    {{1.875f, 3.8125f}, {3.875f, 2.8125f}, {3.6875f, 7.4375f}},
    {{1.25f,  1.625f},  {2.0f,   3.75f},   {4.125f,   2.875f}}};

__device__ __forceinline__ float clampp(float x) {
    return fminf(fmaxf(x, EPSV), 1.0f - EPSV);
}

// ---- Phase 0: zero/one-init the scratch state (harness does not re-poison) ----
__global__ void k_init(int* __restrict__ winner, unsigned char* __restrict__ noobj,
                       float* __restrict__ accs, int* __restrict__ n_pos) {
    int i = blockIdx.x * blockDim.x + threadIdx.x;
    if (i < NB * TOTAL) {
        winner[i] = 0;
        noobj[i] = 1;
    }
    if (i == 0) {
        for (int k = 0; k < 8; ++k) accs[k] = 0.0f;
        *n_pos = 0;
    }
}

// ---- Phase 1: per-target scatter: ignore mask + dedup'd winner per cell ----
__global__ void k_build(const float* __restrict__ targets,
                        int* __restrict__ winner, unsigned char* __restrict__ noobj) {
    int tid = blockIdx.x * blockDim.x + threadIdx.x;
    if (tid >= NB * NT) return;
    int b = tid / NT, n = tid % NT;
    const float* t = targets + (size_t)tid * 5;
    float t0 = t[0], tx = t[1], ty = t[2], tw = t[3], th = t[4];
    if ((t0 + tx + ty + tw + th) == 0.0f) return;  // invalid target

    float best_iou = -1.0f;
    int best_idx = 0;
    for (int m = 0; m < 3; ++m) {
        float fm = c_fm[m];
        float gx = tx * fm, gy = ty * fm, gw = tw * fm, gh = th * fm;
        int gi = (int)gx, gj = (int)gy;
        int base = c_off[m] + 3 * gi * gj;  // matches reference's A*gi*gj indexing
        float liou = -1.0f;
        int la = 0;
        for (int a = 0; a < 3; ++a) {
            float aw = c_anc[m][a][0], ah = c_anc[m][a][1];
            float inter = fminf(gw, aw) * fminf(gh, ah);
            float uni = gw * gh + aw * ah - inter;
            float iou = inter / (uni + 1e-16f);
            if (iou > IGNORE_THR) noobj[b * TOTAL + base + a] = 0;  // races all write 0
            if (iou > liou) { liou = iou; la = a; }                 // first-max wins
        }
        if (liou > best_iou) { best_iou = liou; best_idx = base + la; }
    }
    atomicMax(&winner[b * TOTAL + best_idx], n + 1);  // dedup duplicate cells
}

// ---- Phase 2: sparse positive anchors: xywh MSE, conf-pos BCE, class BCE ----
__global__ void k_pos(const float* __restrict__ pred, const float* __restrict__ targets,
                      const int* __restrict__ winner,
                      float* __restrict__ accs, int* __restrict__ n_pos) {
    int j = blockIdx.x * blockDim.x + threadIdx.x;
    if (j >= NB * TOTAL) return;
    int w = winner[j];
    if (!w) return;
    int b = j / TOTAL;
    const float* t = targets + ((size_t)b * NT + (size_t)(w - 1)) * 5;
    int ci = (int)t[0];
    float txv = t[1] * IMG_SIZE, tyv = t[2] * IMG_SIZE;
    float twv = t[3] * IMG_SIZE, thv = t[4] * IMG_SIZE;
    float wwh = 2.0f - t[3] * t[4];
    float w2 = wwh * wwh;

    const float* p = pred + (size_t)j * PRED_C;
    float px = p[0], py = p[1], pw = p[2], ph = p[3], pc = p[4];
    atomicAdd(&accs[0], w2 * (px - txv) * (px - txv));
    atomicAdd(&accs[1], w2 * (py - tyv) * (py - tyv));
    atomicAdd(&accs[2], w2 * (pw - twv) * (pw - twv));
    atomicAdd(&accs[3], w2 * (ph - thv) * (ph - thv));
    atomicAdd(&accs[4], -logf(clampp(pc)));  // tconf==1 positive BCE term

    float s = 0.0f;
    for (int c = 0; c < NUM_CLASSES; ++c) {
        float q = clampp(p[5 + c]);
        s += (c == ci) ? -logf(q) : -logf(1.0f - q);
    }
    atomicAdd(&accs[5], s);
    atomicAdd(n_pos, 1);
}

// ---- Phase 3: dense no-object conf BCE, reduced on the WMMA pipe ----
// D = ones(16x4) x B(4x16) + C : every B element contributes once to the
// row-replicated column sums; sum(acc[0] over 32 lanes) == 2 * sum(values).
// Branchless body: mask byte + conf loaded unconditionally (tail indices
// clamped), result selected -- both loads issue before one wait, no EXEC churn.
__global__ void k_noobj(const float* __restrict__ pred,
                        const unsigned char* __restrict__ noobj,
                        float* __restrict__ accs) {
    const long NPOS = (long)NB * TOTAL;
    int gtid = blockIdx.x * blockDim.x + threadIdx.x;
    int wave = gtid >> 5, lane = gtid & 31;
    long nwaves = (long)(gridDim.x * blockDim.x) >> 5;
    long stride = nwaves * 64;

    const float cneg = -logf(1.0f - EPSV);  // term at noobj==0: -log(1-clip(0))
    v2f a = {1.0f, 1.0f};                   // all-ones 16x4 A operand
    v8f acc = {};

    for (long base = (long)wave * 64; base < NPOS; base += stride) {
        long j0 = base + lane, j1 = base + 32 + lane;
        long i0 = j0 < NPOS - 1 ? j0 : NPOS - 1;  // clamp tail, load always
        long i1 = j1 < NPOS - 1 ? j1 : NPOS - 1;

        unsigned char m0 = noobj[i0];
        unsigned char m1 = noobj[i1];
        float c0 = pred[i0 * PRED_C + 4];
        float c1 = pred[i1 * PRED_C + 4];
        __builtin_prefetch(&pred[i0 * PRED_C + 4 + stride * PRED_C], 0, 3);

        float t0 = m0 ? -logf(1.0f - clampp(c0)) : cneg;
        float t1 = m1 ? -logf(1.0f - clampp(c1)) : cneg;
        float v0 = (j0 < NPOS) ? t0 : 0.0f;
        float v1 = (j1 < NPOS) ? t1 : 0.0f;

        v2f bb = {v0, v1};
        acc = __builtin_amdgcn_wmma_f32_16x16x4_f32(
            /*neg_a=*/false, a, /*neg_b=*/false, bb,
            /*c_mod=*/(short)0, acc, /*reuse_a=*/false, /*reuse_b=*/false);
    }

    float v = acc[0];
    for (int o = 16; o; o >>= 1) v += __shfl_xor(v, o, 32);
    if (lane == 0) atomicAdd(&accs[6], v * 0.5f);
}

// ---- Phase 4: combine partial sums into the scalar loss ----
__global__ void k_final(const float* __restrict__ accs, const int* __restrict__ n_pos,
                        float* __restrict__ out) {
    const float BT = (float)(NB * TOTAL);
    float np = (float)(*n_pos);
    float cneg = -logf(1.0f - EPSV);
    float lxywh = (accs[0] + accs[1] + accs[2] + accs[3]) / BT;
    float lconf = (accs[4] + (BT - np) * cneg) / BT + 0.5f * accs[6] / BT;
    float lcls = accs[5] / (fmaxf(np, 1.0f) * (float)NUM_CLASSES);
    out[0] = lxywh + lconf + lcls;
}

extern "C" void kernel_launch(void* const* d_in, const int* in_sizes, int n_in,
                              void* d_out, int out_size, void* d_ws, size_t ws_size,
                              hipStream_t stream) {
    const float* pred = (const float*)d_in[0];     // (64, 10647, 85) f32
    const float* targets = (const float*)d_in[1];  // (64, 50, 5) f32
    float* out = (float*)d_out;

    // Workspace layout:
    //   [0,32)    : float accs[8] {sx,sy,sw,sh,conf_pos,cls_sum,noobj_sum,-}
    //   [32,36)   : int n_pos
    //   [64, ...) : int winner[NB*TOTAL]   (~2.73 MB)
    //   then      : u8 noobj[NB*TOTAL]     (~0.68 MB)
    float* accs = (float*)d_ws;
    int* n_pos = (int*)((char*)d_ws + 32);
    int* winner = (int*)((char*)d_ws + 64);
    unsigned char* noobj = (unsigned char*)((char*)d_ws + 64 + sizeof(int) * NB * TOTAL);

    const int NPOS = NB * TOTAL;
    k_init<<<(NPOS + 255) / 256, 256, 0, stream>>>(winner, noobj, accs, n_pos);
    k_build<<<(NB * NT + 255) / 256, 256, 0, stream>>>(targets, winner, noobj);
    k_pos<<<(NPOS + 255) / 256, 256, 0, stream>>>(pred, targets, winner, accs, n_pos);
    k_noobj<<<416, 256, 0, stream>>>(pred, noobj, accs);  // 3328 waves, grid-stride
    k_final<<<1, 1, 0, stream>>>(accs, n_pos, out);
}